// EmgLSTMNet_71760313582333
// MI455X (gfx1250) — compile-verified
//
#include <hip/hip_runtime.h>
#include <hip/hip_bf16.h>

typedef float v2f __attribute__((ext_vector_type(2)));
typedef float v8f __attribute__((ext_vector_type(8)));

#define BATCH 256
#define SEQT  2048
#define DIN   16
#define H1    5
#define H2    50
#define G1    20      // 4*H1
#define G2    200     // 4*H2
#define NC    20

#define BT    16      // batch tile (WMMA M)
#define NT2   13      // layer-2 gate N tiles (13*16 = 208 >= 200)
#define KC2H  13      // K-chunks over H2 (pad 50 -> 52, 4 per chunk)
#define KC2I  2       // K-chunks over H1 (pad 5 -> 8)
#define NW    13      // one wave per layer-2 N tile
#define NTHR  (NW*32)

#define LOG2E 1.4426950408889634f

// Branch-free activations: raw v_exp_f32 / v_rcp_f32, no libm range-reduction
// branches on the serial recurrence path. Saturate correctly at +/-inf.
__device__ __forceinline__ float fexp(float x)  { return __builtin_amdgcn_exp2f(x * LOG2E); }
__device__ __forceinline__ float frcp(float x)  { return __builtin_amdgcn_rcpf(x); }
__device__ __forceinline__ float sigm(float x)  { return frcp(1.0f + fexp(-x)); }
__device__ __forceinline__ float tanhx(float x) { return 1.0f - 2.0f * frcp(fexp(2.0f * x) + 1.0f); }

__device__ __forceinline__ v8f wmma_f32(v2f a, v2f b, v8f c) {
    // V_WMMA_F32_16X16X4_F32 : D = A(16x4,f32) * B(4x16,f32) + C(16x16,f32)
    return __builtin_amdgcn_wmma_f32_16x16x4_f32(false, a, false, b, (short)0, c, false, false);
}

__global__ __launch_bounds__(NTHR) void lstm2_fused_kernel(
    const float* __restrict__ x,
    const float* __restrict__ W1ih, const float* __restrict__ W1hh,
    const float* __restrict__ b1ih, const float* __restrict__ b1hh,
    const float* __restrict__ W2ih, const float* __restrict__ W2hh,
    const float* __restrict__ b2ih, const float* __restrict__ b2hh,
    const float* __restrict__ Wfc,  const float* __restrict__ bfc,
    float* __restrict__ out)
{
    __shared__ float h1s[BT][8];          // layer-1 hidden state (pad 5->8, pads stay 0)
    __shared__ float h2s[BT][52];         // layer-2 hidden state (pad 50->52, pads stay 0)
    __shared__ float g1s[BT][32];         // layer-1 gate tile
    __shared__ float g2s[BT][NT2*16];     // layer-2 gate tile (208 cols)
    __shared__ float b1s[32];             // b1ih+b1hh (padded)
    __shared__ float b2s[NT2*16];         // b2ih+b2hh (padded)

    const int tid   = threadIdx.x;
    const int w     = tid >> 5;           // wave id (wave32)
    const int lane  = tid & 31;
    const int nloc  = lane & 15;          // A-row (M) / B,C,D-column (N) within tile
    const int khalf = (lane >> 4) * 2;    // K sub-offset per half-wave (ISA 16x4 layout)
    const int b0    = blockIdx.x * BT;

    // ---- zero recurrent state, precompute fused biases ----
    for (int i = tid; i < BT*8;     i += NTHR) h1s[i >> 3][i & 7]   = 0.0f;
    for (int i = tid; i < BT*52;    i += NTHR) h2s[i / 52][i % 52]  = 0.0f;
    for (int i = tid; i < 32;       i += NTHR) b1s[i] = (i < G1) ? (b1ih[i] + b1hh[i]) : 0.0f;
    for (int i = tid; i < NT2*16;   i += NTHR) b2s[i] = (i < G2) ? (b2ih[i] + b2hh[i]) : 0.0f;

    // ---- register-resident B fragments: layer-2 weights (one N tile per wave) ----
    v2f whh[KC2H], wih[KC2I];
    {
        const int ng = w * 16 + nloc;     // gate column this lane supplies
        #pragma unroll
        for (int kc = 0; kc < KC2H; ++kc) {
            const int k0 = kc * 4 + khalf;
            whh[kc].x = (ng < G2 && k0     < H2) ? W2hh[ng * H2 + k0]     : 0.0f;
            whh[kc].y = (ng < G2 && k0 + 1 < H2) ? W2hh[ng * H2 + k0 + 1] : 0.0f;
        }
        #pragma unroll
        for (int kc = 0; kc < KC2I; ++kc) {
            const int k0 = kc * 4 + khalf;
            wih[kc].x = (ng < G2 && k0     < H1) ? W2ih[ng * H1 + k0]     : 0.0f;
            wih[kc].y = (ng < G2 && k0 + 1 < H1) ? W2ih[ng * H1 + k0 + 1] : 0.0f;
        }
    }
    // ---- layer-1 weight fragments (used by waves 0,1) ----
    v2f w1i[4], w1h[2];
    {
        const bool act = (w < 2);
        const int  ng  = (w & 1) * 16 + nloc;
        #pragma unroll
        for (int kc = 0; kc < 4; ++kc) {
            const int k0 = kc * 4 + khalf;
            w1i[kc].x = (act && ng < G1) ? W1ih[ng * DIN + k0]     : 0.0f;
            w1i[kc].y = (act && ng < G1) ? W1ih[ng * DIN + k0 + 1] : 0.0f;
        }
        #pragma unroll
        for (int kc = 0; kc < 2; ++kc) {
            const int k0 = kc * 4 + khalf;
            w1h[kc].x = (act && ng < G1 && k0     < H1) ? W1hh[ng * H1 + k0]     : 0.0f;
            w1h[kc].y = (act && ng < G1 && k0 + 1 < H1) ? W1hh[ng * H1 + k0 + 1] : 0.0f;
        }
    }

    float c1p = 0.0f;                 // private cell state, layer 1 (threads < 80)
    float c2a = 0.0f, c2b = 0.0f;     // private cell state, layer 2 (<=2 elems/thread)

    const float* xrow = x + (size_t)(b0 + nloc) * SEQT * DIN;   // A-row base for this lane

    __syncthreads();

    for (int t = 0; t < SEQT; ++t) {
        // ======== layer-1 gates: (16x21) @ (21x20), waves 0,1 ========
        if (w < 2) {
            v8f acc = {0.f,0.f,0.f,0.f,0.f,0.f,0.f,0.f};
            const float* xa = xrow + (size_t)t * DIN + khalf;
            if (t + 16 < SEQT) __builtin_prefetch(xa + 16 * DIN, 0, 3);
            #pragma unroll
            for (int kc = 0; kc < 4; ++kc) {                    // K over x (16)
                v2f a = *(const v2f*)(xa + kc * 4);
                acc = wmma_f32(a, w1i[kc], acc);
            }
            #pragma unroll
            for (int kc = 0; kc < 2; ++kc) {                    // K over h1 (pad 8)
                v2f a = *(const v2f*)(&h1s[nloc][kc * 4 + khalf]);
                acc = wmma_f32(a, w1h[kc], acc);
            }
            #pragma unroll
            for (int r = 0; r < 8; ++r)                         // D tile -> LDS
                g1s[r + 8 * (lane >> 4)][(w & 1) * 16 + nloc] = acc[r];
        }
        __syncthreads();

        // ======== layer-1 pointwise: 16x5 elements ========
        if (tid < BT * H1) {
            const int bb = tid / H1, j = tid % H1;
            const float ig = sigm (g1s[bb][j]          + b1s[j]);
            const float fg = sigm (g1s[bb][H1 + j]     + b1s[H1 + j]);
            const float gg = tanhx(g1s[bb][2 * H1 + j] + b1s[2 * H1 + j]);
            const float og = sigm (g1s[bb][3 * H1 + j] + b1s[3 * H1 + j]);
            c1p = fg * c1p + ig * gg;
            h1s[bb][j] = og * tanhx(c1p);
        }
        __syncthreads();

        // ======== layer-2 gates: (16x55) @ (55x200), one N tile per wave ========
        {
            v8f acc = {0.f,0.f,0.f,0.f,0.f,0.f,0.f,0.f};
            #pragma unroll
            for (int kc = 0; kc < KC2I; ++kc) {                 // K over h1 (pad 8)
                v2f a = *(const v2f*)(&h1s[nloc][kc * 4 + khalf]);
                acc = wmma_f32(a, wih[kc], acc);
            }
            #pragma unroll
            for (int kc = 0; kc < KC2H; ++kc) {                 // K over h2 (pad 52)
                v2f a = *(const v2f*)(&h2s[nloc][kc * 4 + khalf]);
                acc = wmma_f32(a, whh[kc], acc);
            }
            #pragma unroll
            for (int r = 0; r < 8; ++r)
                g2s[r + 8 * (lane >> 4)][w * 16 + nloc] = acc[r];
        }
        __syncthreads();

        // ======== layer-2 pointwise: 16x50 = 800 elements over 416 threads ========
        {
            const int bb = tid / H2, j = tid % H2;              // tid < 800 always here
            const float ig = sigm (g2s[bb][j]          + b2s[j]);
            const float fg = sigm (g2s[bb][H2 + j]     + b2s[H2 + j]);
            const float gg = tanhx(g2s[bb][2 * H2 + j] + b2s[2 * H2 + j]);
            const float og = sigm (g2s[bb][3 * H2 + j] + b2s[3 * H2 + j]);
            c2a = fg * c2a + ig * gg;
            h2s[bb][j] = og * tanhx(c2a);
        }
        {
            const int e = tid + NTHR;
            if (e < BT * H2) {
                const int bb = e / H2, j = e % H2;
                const float ig = sigm (g2s[bb][j]          + b2s[j]);
                const float fg = sigm (g2s[bb][H2 + j]     + b2s[H2 + j]);
                const float gg = tanhx(g2s[bb][2 * H2 + j] + b2s[2 * H2 + j]);
                const float og = sigm (g2s[bb][3 * H2 + j] + b2s[3 * H2 + j]);
                c2b = fg * c2b + ig * gg;
                h2s[bb][j] = og * tanhx(c2b);
            }
        }
        __syncthreads();
    }

    // ======== outputs: last = h2_{T-1}; result = last @ Wfc.T + bfc ========
    float* out_res  = out;                       // [256][20]
    float* out_last = out + (size_t)BATCH * NC;  // [256][50]
    for (int e = tid; e < BT * H2; e += NTHR) {
        const int bb = e / H2, j = e % H2;
        out_last[(size_t)(b0 + bb) * H2 + j] = h2s[bb][j];
    }
    if (w < 2) {
        const int ng = w * 16 + nloc;
        v8f acc = {0.f,0.f,0.f,0.f,0.f,0.f,0.f,0.f};
        #pragma unroll
        for (int kc = 0; kc < KC2H; ++kc) {
            const int k0 = kc * 4 + khalf;
            v2f bf;
            bf.x = (ng < NC && k0     < H2) ? Wfc[ng * H2 + k0]     : 0.0f;
            bf.y = (ng < NC && k0 + 1 < H2) ? Wfc[ng * H2 + k0 + 1] : 0.0f;
            v2f a = *(const v2f*)(&h2s[nloc][k0]);
            acc = wmma_f32(a, bf, acc);
        }
        #pragma unroll
        for (int r = 0; r < 8; ++r) {
            const int m = r + 8 * (lane >> 4);
            if (ng < NC)
                out_res[(size_t)(b0 + m) * NC + ng] = acc[r] + bfc[ng];
        }
    }
}

extern "C" void kernel_launch(void* const* d_in, const int* in_sizes, int n_in,
                              void* d_out, int out_size, void* d_ws, size_t ws_size,
                              hipStream_t stream) {
    (void)in_sizes; (void)n_in; (void)out_size; (void)d_ws; (void)ws_size;
    const float* x    = (const float*)d_in[0];
    const float* W1ih = (const float*)d_in[1];
    const float* W1hh = (const float*)d_in[2];
    const float* b1ih = (const float*)d_in[3];
    const float* b1hh = (const float*)d_in[4];
    const float* W2ih = (const float*)d_in[5];
    const float* W2hh = (const float*)d_in[6];
    const float* b2ih = (const float*)d_in[7];
    const float* b2hh = (const float*)d_in[8];
    const float* Wfc  = (const float*)d_in[9];
    const float* bfc  = (const float*)d_in[10];
    float* out = (float*)d_out;

    dim3 grid(BATCH / BT);     // 16 workgroups, one per batch tile
    dim3 block(NTHR);          // 13 waves (wave32)
    lstm2_fused_kernel<<<grid, block, 0, stream>>>(
        x, W1ih, W1hh, b1ih, b1hh, W2ih, W2hh, b2ih, b2hh, Wfc, bfc, out);
}